// MIL_23270132809746
// MI455X (gfx1250) — compile-verified
//
#include <hip/hip_runtime.h>
#include <hip/hip_bf16.h>
#include <math.h>

// ---------------- problem constants (match reference) ----------------
constexpr int   Cc   = 512;
constexpr int   N1   = 8192;
constexpr int   N2   = 131072;
constexpr int   Nn   = 1 + N1 + N2;          // 139265
constexpr int   Ee   = 1000000;
constexpr int   STEP = 10;
constexpr int   K1   = (N1 + STEP - 1) / STEP;  // 820
constexpr int   K2   = 4 * K1;                   // 3280
constexpr int   Mm   = 1 + K1 + K2;              // 4101

// ---------------- output layout (flat f32, in return order) ----------
constexpr size_t OFF_XPOOL   = 0;
constexpr size_t OFF_A       = OFF_XPOOL + (size_t)Mm * Cc;
constexpr size_t OFF_BATCH   = OFF_A + (size_t)Mm * Mm;
constexpr size_t OFF_CLUSTER = OFF_BATCH + Mm;
constexpr size_t OFF_NNT     = OFF_CLUSTER + Nn;
constexpr size_t OFF_NTREE   = OFF_NNT + Mm;
constexpr size_t OFF_FIT     = OFF_NTREE + Mm;
constexpr size_t OFF_NXY     = OFF_FIT + Nn;
constexpr size_t OUT_TOTAL   = OFF_NXY + 2 * (size_t)Mm;

// ---------------- workspace layout (4-byte units) --------------------
constexpr size_t WS_NORM   = 0;                 // 2 floats
constexpr size_t WS_F1     = 2;                 // N1 floats
constexpr size_t WS_F2     = WS_F1 + N1;        // N2 floats
constexpr size_t WS_CNT    = WS_F2 + N2;        // Mm floats
constexpr size_t WS_ORDER1 = WS_CNT + Mm;       // N1 ints
constexpr size_t WS_CL1    = WS_ORDER1 + N1;    // N1 ints
constexpr size_t WS_TMIN   = WS_CL1 + N1;       // 1 int
constexpr size_t WS_PAR    = WS_TMIN + 1;       // N2 ints
constexpr size_t WS_COUNTS = WS_PAR + N2;       // K1 ints
constexpr size_t WS_STARTS = WS_COUNTS + K1;    // K1 ints
constexpr size_t WS_CURSOR = WS_STARTS + K1;    // K1 ints
constexpr size_t WS_BUCKET = WS_CURSOR + K1;    // N2 ints
constexpr size_t WS_CAND   = WS_BUCKET + N2;    // 4*K1 ints
constexpr size_t WS_CL2    = WS_CAND + 4 * K1;  // N2 ints
constexpr size_t WS_CLUST  = WS_CL2 + N2;       // Nn ints

typedef float v2f __attribute__((ext_vector_type(2)));
typedef float v8f __attribute__((ext_vector_type(8)));

__device__ __forceinline__ float finf() { return __builtin_inff(); }

// ---------------- utility kernels ----------------
__global__ void k_zero_f(float* p, size_t n) {
    size_t i = (size_t)blockIdx.x * blockDim.x + threadIdx.x;
    size_t stride = (size_t)gridDim.x * blockDim.x;
    for (; i < n; i += stride) p[i] = 0.0f;
}

__global__ void k_init_small(int* counts3, int* tmin) {
    int i = blockIdx.x * blockDim.x + threadIdx.x;
    if (i < 3 * K1) counts3[i] = 0;
    if (i == 0) *tmin = 0x7fffffff;
}

// ---------------- ||w1||, ||w2|| ----------------
__global__ void k_norms(const float* __restrict__ w1, const float* __restrict__ w2,
                        float* __restrict__ norms) {
    __shared__ float s[2];
    if (threadIdx.x == 0) { s[0] = 0.f; s[1] = 0.f; }
    __syncthreads();
    float a = 0.f, b = 0.f;
    for (int i = threadIdx.x; i < Cc; i += blockDim.x) {
        a += w1[i] * w1[i];
        b += w2[i] * w2[i];
    }
    atomicAdd(&s[0], a);
    atomicAdd(&s[1], b);
    __syncthreads();
    if (threadIdx.x == 0) { norms[0] = sqrtf(s[0]); norms[1] = sqrtf(s[1]); }
}

// ---------------- f = tanh((x . w)/||w||) via V_WMMA_F32_16X16X4_F32 ----
// One wave computes a 16-row tile: D(16x16) += A(16x4) * B(4x16), where all 16
// columns of B carry the same 4-element chunk of w, so D[:,0] = partial dots.
// A layout (16x4 f32): lanes 0-15 hold M=lane K=0..1, lanes 16-31 hold M=lane-16 K=2..3.
// C/D layout: vgpr v, lanes 0-15 -> M=v, lanes 16-31 -> M=v+8.
__global__ void k_fitness_wmma(const float* __restrict__ x, const float* __restrict__ w,
                               const float* __restrict__ norms, int which,
                               float* __restrict__ f, int nrows) {
    __shared__ float wl[Cc];
    for (int i = threadIdx.x; i < Cc; i += blockDim.x) wl[i] = w[i];
    __syncthreads();
    const float nrm = norms[which];

    const int wave = threadIdx.x >> 5;
    const int lane = threadIdx.x & 31;
    const int tile = blockIdx.x * (blockDim.x >> 5) + wave;
    const int row0 = tile * 16;
    if (row0 >= nrows) return;  // wave-uniform

    const int m    = lane & 15;
    const int half = lane >> 4;     // 0: K=0,1  1: K=2,3
    const float* xr = x + (size_t)(row0 + m) * Cc + half * 2;
    const float* wr = wl + half * 2;

    v8f acc = {0.f, 0.f, 0.f, 0.f, 0.f, 0.f, 0.f, 0.f};
    #pragma unroll 8
    for (int k0 = 0; k0 < Cc; k0 += 4) {
        v2f a; a.x = xr[k0];  a.y = xr[k0 + 1];
        v2f b; b.x = wr[k0];  b.y = wr[k0 + 1];
        acc = __builtin_amdgcn_wmma_f32_16x16x4_f32(
            /*neg_a=*/false, a, /*neg_b=*/false, b,
            /*c_mod=*/(short)0, acc, /*reuse_a=*/false, /*reuse_b=*/false);
    }
    if (lane == 0) {
        #pragma unroll
        for (int v = 0; v < 8; ++v) f[row0 + v] = tanhf(acc[v] / nrm);
    } else if (lane == 16) {
        #pragma unroll
        for (int v = 0; v < 8; ++v) f[row0 + 8 + v] = tanhf(acc[v] / nrm);
    }
}

// ---------------- stable argsort of f1 via rank counting ----------------
__global__ void k_rank_order(const float* __restrict__ f, int n, int* __restrict__ order) {
    int j = blockIdx.x * blockDim.x + threadIdx.x;
    if (j >= n) return;
    const float fj = f[j];
    int r = 0;
    for (int i = 0; i < n; ++i) {
        float fi = f[i];
        r += (fi < fj) || (fi == fj && i < j);
    }
    order[r] = j;
}

// ---------------- cluster_1: argmin over K1 threshold rows ----------------
__global__ void k_cluster1(const float* __restrict__ xy, const float* __restrict__ f1,
                           const int* __restrict__ order1, int* __restrict__ cl1) {
    __shared__ float tx[K1], ty[K1], tf[K1];
    for (int t = threadIdx.x; t < K1; t += blockDim.x) {
        int j = order1[t * STEP];
        tx[t] = xy[2 * (1 + j)];
        ty[t] = xy[2 * (1 + j) + 1];
        tf[t] = f1[j];
    }
    __syncthreads();
    int j = blockIdx.x * blockDim.x + threadIdx.x;
    if (j >= N1) return;
    const float px = xy[2 * (1 + j)], py = xy[2 * (1 + j) + 1], pf = f1[j];
    float best = finf();
    int bi = 0;
    for (int k = 0; k < K1; ++k) {
        float dx = tx[k] - px, dy = ty[k] - py;
        float d = sqrtf(dx * dx + dy * dy) + fabsf(tf[k] - pf);
        if (d < best) { best = d; bi = k; }
    }
    cl1[j] = bi;
}

// ---------------- min(tree2) ----------------
__global__ void k_treemin(const int* __restrict__ tree, int* __restrict__ tmin) {
    int i = blockIdx.x * blockDim.x + threadIdx.x;
    if (i < N2) atomicMin(tmin, tree[1 + N1 + i]);
}

// ---------------- parents + counts ----------------
__global__ void k_parents(const int* __restrict__ tree, const int* __restrict__ tmin,
                          const int* __restrict__ cl1, int* __restrict__ parents,
                          int* __restrict__ counts) {
    int i = blockIdx.x * blockDim.x + threadIdx.x;
    if (i >= N2) return;
    int p = cl1[tree[1 + N1 + i] - *tmin];
    parents[i] = p;
    atomicAdd(&counts[p], 1);
}

__global__ void k_scan(const int* __restrict__ counts, int* __restrict__ starts) {
    if (blockIdx.x == 0 && threadIdx.x == 0) {
        int s = 0;
        for (int p = 0; p < K1; ++p) { starts[p] = s; s += counts[p]; }
    }
}

__global__ void k_scatter(const int* __restrict__ parents, const int* __restrict__ starts,
                          int* __restrict__ cursor, int* __restrict__ bucket) {
    int i = blockIdx.x * blockDim.x + threadIdx.x;
    if (i >= N2) return;
    int p = parents[i];
    int pos = starts[p] + atomicAdd(&cursor[p], 1);
    bucket[pos] = i;
}

// ---------------- top-4 smallest (comp_key, idx) per parent ----------------
// comp_key = 4.0f*p + f2 to reproduce the reference's float key (incl. rounding);
// index tiebreak matches stable argsort.
__global__ void k_cand(const int* __restrict__ counts, const int* __restrict__ starts,
                       const int* __restrict__ bucket, const float* __restrict__ f2,
                       int* __restrict__ cand) {
    int p = blockIdx.x * blockDim.x + threadIdx.x;
    if (p >= K1) return;
    int s = starts[p], c = counts[p];
    float bk[4] = {finf(), finf(), finf(), finf()};
    int   bi[4] = {-1, -1, -1, -1};
    for (int t = 0; t < c; ++t) {
        int i = bucket[s + t];
        float key = 4.0f * (float)p + f2[i];
        #pragma unroll
        for (int r = 0; r < 4; ++r) {
            bool better = (bi[r] < 0) || (key < bk[r]) || (key == bk[r] && i < bi[r]);
            if (better) {
                for (int q = 3; q > r; --q) { bk[q] = bk[q - 1]; bi[q] = bi[q - 1]; }
                bk[r] = key; bi[r] = i;
                break;
            }
        }
    }
    #pragma unroll
    for (int r = 0; r < 4; ++r) cand[4 * p + r] = bi[r];
}

// ---------------- cluster_2 ----------------
__global__ void k_cluster2(const int* __restrict__ parents, const int* __restrict__ cand,
                           const float* __restrict__ xy, const float* __restrict__ f2,
                           int* __restrict__ cl2) {
    int i = blockIdx.x * blockDim.x + threadIdx.x;
    if (i >= N2) return;
    int p = parents[i];
    const float px = xy[2 * (1 + N1 + i)], py = xy[2 * (1 + N1 + i) + 1], pf = f2[i];
    float best = finf();
    int loc = 0;
    #pragma unroll
    for (int r = 0; r < 4; ++r) {
        int c = cand[4 * p + r];
        float d;
        if (c < 0) {
            d = finf();
        } else {
            float ax = xy[2 * (1 + N1 + c)], ay = xy[2 * (1 + N1 + c) + 1], af = f2[c];
            float dx = ax - px, dy = ay - py;
            d = sqrtf(dx * dx + dy * dy) + fabsf(af - pf);
        }
        if (d < best) { best = d; loc = r; }
    }
    cl2[i] = 4 * p + loc;
}

// ---------------- assemble cluster / fitness / cnt / A-diagonal ----------------
__global__ void k_assemble(const int* __restrict__ cl1, const int* __restrict__ cl2,
                           const float* __restrict__ f1, const float* __restrict__ f2,
                           int* __restrict__ cluster, float* __restrict__ outCl,
                           float* __restrict__ outFit, float* __restrict__ cnt,
                           float* __restrict__ Amat) {
    int i = blockIdx.x * blockDim.x + threadIdx.x;
    if (i >= Nn) return;
    int cl;
    float fit;
    if (i == 0)        { cl = 0;                       fit = 0.0f; }
    else if (i <= N1)  { cl = cl1[i - 1] + 1;          fit = f1[i - 1]; }
    else               { cl = cl2[i - 1 - N1] + 1 + K1; fit = f2[i - 1 - N1]; }
    cluster[i] = cl;
    outCl[i] = (float)cl;
    outFit[i] = fit;
    atomicAdd(&cnt[cl], 1.0f);
    atomicAdd(&Amat[(size_t)cl * Mm + cl], 1.0f);   // self-loop edge
}

// ---------------- metadata outputs ----------------
__global__ void k_meta(float* __restrict__ nnt, float* __restrict__ ntree) {
    int m = blockIdx.x * blockDim.x + threadIdx.x;
    if (m >= Mm) return;
    nnt[m]   = (m == 0) ? 0.0f : (m <= K1 ? 1.0f : 2.0f);
    ntree[m] = (m == 0) ? -1.0f : (m <= K1 ? 0.0f : (float)(1 + (m - 1 - K1) / 4));
}

// ---------------- pooled sums (x, xy) ----------------
__global__ void k_pool(const float* __restrict__ x, const float* __restrict__ xy,
                       const int* __restrict__ cluster, float* __restrict__ xpool,
                       float* __restrict__ nxy) {
    int n = blockIdx.x;
    int cl = cluster[n];
    const float* xr = x + (size_t)n * Cc;
    float* dst = xpool + (size_t)cl * Cc;
    for (int c = threadIdx.x; c < Cc; c += blockDim.x) atomicAdd(&dst[c], xr[c]);
    if (threadIdx.x < 2) atomicAdd(&nxy[2 * cl + threadIdx.x], xy[2 * n + threadIdx.x]);
}

__global__ void k_divide(float* __restrict__ v, const float* __restrict__ cnt,
                         size_t n, int stride) {
    size_t i = (size_t)blockIdx.x * blockDim.x + threadIdx.x;
    if (i >= n) return;
    v[i] /= fmaxf(cnt[i / stride], 1.0f);
}

// ---------------- edge histogram into A ----------------
__global__ void k_edges(const int* __restrict__ ei, const int* __restrict__ cluster,
                        float* __restrict__ Amat) {
    int i = blockIdx.x * blockDim.x + threadIdx.x;
    if (i >= Ee) return;
    int r = cluster[ei[i]];
    int c = cluster[ei[Ee + i]];
    atomicAdd(&Amat[(size_t)r * Mm + c], 1.0f);
}

// =====================================================================
extern "C" void kernel_launch(void* const* d_in, const int* in_sizes, int n_in,
                              void* d_out, int out_size, void* d_ws, size_t ws_size,
                              hipStream_t stream) {
    const float* x    = (const float*)d_in[0];
    const int*   ei   = (const int*)d_in[1];
    // d_in[2] = node_type (unused by the reference computation)
    const int*   tree = (const int*)d_in[3];
    const float* xy   = (const float*)d_in[4];
    const float* w1   = (const float*)d_in[5];
    const float* w2   = (const float*)d_in[6];

    float* out  = (float*)d_out;
    float* wsf  = (float*)d_ws;
    int*   wsi  = (int*)d_ws;

    float* norms   = wsf + WS_NORM;
    float* f1      = wsf + WS_F1;
    float* f2      = wsf + WS_F2;
    float* cnt     = wsf + WS_CNT;
    int*   order1  = wsi + WS_ORDER1;
    int*   cl1     = wsi + WS_CL1;
    int*   tmin    = wsi + WS_TMIN;
    int*   parents = wsi + WS_PAR;
    int*   counts  = wsi + WS_COUNTS;
    int*   starts  = wsi + WS_STARTS;
    int*   cursor  = wsi + WS_CURSOR;
    int*   bucket  = wsi + WS_BUCKET;
    int*   cand    = wsi + WS_CAND;
    int*   cl2     = wsi + WS_CL2;
    int*   cluster = wsi + WS_CLUST;

    const int TB = 256;

    // zero all accumulated outputs + cnt, init counts/starts/cursor/tmin
    k_zero_f<<<4096, TB, 0, stream>>>(out, OUT_TOTAL);
    k_zero_f<<<64, TB, 0, stream>>>(cnt, (size_t)Mm);
    k_init_small<<<(3 * K1 + TB - 1) / TB, TB, 0, stream>>>(counts, tmin);

    // weight norms
    k_norms<<<1, TB, 0, stream>>>(w1, w2, norms);

    // fitness via f32 WMMA (16 rows per wave, 8 waves per block => 128 rows/block)
    k_fitness_wmma<<<N1 / 128, TB, 0, stream>>>(x + (size_t)Cc, w1, norms, 0, f1, N1);
    k_fitness_wmma<<<N2 / 128, TB, 0, stream>>>(x + (size_t)Cc * (1 + N1), w2, norms, 1, f2, N2);

    // stable argsort of f1, threshold rows, cluster_1
    k_rank_order<<<(N1 + TB - 1) / TB, TB, 0, stream>>>(f1, N1, order1);
    k_cluster1<<<(N1 + TB - 1) / TB, TB, 0, stream>>>(xy, f1, order1, cl1);

    // parents, bucketing, top-4 candidates, cluster_2
    k_treemin<<<(N2 + TB - 1) / TB, TB, 0, stream>>>(tree, tmin);
    k_parents<<<(N2 + TB - 1) / TB, TB, 0, stream>>>(tree, tmin, cl1, parents, counts);
    k_scan<<<1, 1, 0, stream>>>(counts, starts);
    k_scatter<<<(N2 + TB - 1) / TB, TB, 0, stream>>>(parents, starts, cursor, bucket);
    k_cand<<<(K1 + TB - 1) / TB, TB, 0, stream>>>(counts, starts, bucket, f2, cand);
    k_cluster2<<<(N2 + TB - 1) / TB, TB, 0, stream>>>(parents, cand, xy, f2, cl2);

    // assemble cluster/fitness/cnt/self-loops, metadata
    k_assemble<<<(Nn + TB - 1) / TB, TB, 0, stream>>>(cl1, cl2, f1, f2, cluster,
                                                      out + OFF_CLUSTER, out + OFF_FIT,
                                                      cnt, out + OFF_A);
    k_meta<<<(Mm + TB - 1) / TB, TB, 0, stream>>>(out + OFF_NNT, out + OFF_NTREE);

    // segment sums -> means
    k_pool<<<Nn, TB, 0, stream>>>(x, xy, cluster, out + OFF_XPOOL, out + OFF_NXY);
    k_divide<<<(int)(((size_t)Mm * Cc + TB - 1) / TB), TB, 0, stream>>>(out + OFF_XPOOL, cnt,
                                                                        (size_t)Mm * Cc, Cc);
    k_divide<<<(2 * Mm + TB - 1) / TB, TB, 0, stream>>>(out + OFF_NXY, cnt, (size_t)2 * Mm, 2);

    // adjacency histogram (batch output stays zero from k_zero_f)
    k_edges<<<(Ee + TB - 1) / TB, TB, 0, stream>>>(ei, cluster, out + OFF_A);
}